// DecorelationNormalization_30279519437700
// MI455X (gfx1250) — compile-verified
//
#include <hip/hip_runtime.h>
#include <hip/hip_bf16.h>

// IterNorm (decorrelation whitening) for x:(64,56,56,256) f32, groups of 64.
// Pipeline: [cov+sums] -> [Newton-Schulz inverse-sqrt per group] -> [whiten+affine]
// Heavy matmuls via V_WMMA_F32_16X16X4_F32 (wave32, CDNA5).
// cov kernel uses a pair-interleaved LDS tile so every WMMA operand is one
// conflict-free ds_load_b64; both streaming kernels software-pipeline global
// loads against WMMA compute.

typedef __attribute__((ext_vector_type(2))) float v2f;
typedef __attribute__((ext_vector_type(8))) float v8f;

#define EPSV 1e-5f
#define TPB 32          // p-rows per LDS tile in cov kernel
#define STRW 544        // words per p-PAIR row in interleaved cov tile (512+32)
#define LDSC 68         // padded row stride for 64-wide matrices

__device__ __forceinline__ v8f wmma_f32(v2f a, v2f b, v8f c) {
    return __builtin_amdgcn_wmma_f32_16x16x4_f32(
        /*neg_a=*/false, a, /*neg_b=*/false, b,
        /*c_mod=*/(short)0, c, /*reuse_a=*/false, /*reuse_b=*/false);
}

// ---------------------------------------------------------------------------
// Kernel 1: per-group Gram matrices S[g] = sum_p x_p x_p^T and per-channel sums
// LDS layout: element (p,c) at word (p>>1)*STRW + 2c + (p&1); K-pairs (p0,p0+1)
// with even p0 are contiguous -> single b64 per WMMA operand.
// ---------------------------------------------------------------------------
__global__ __launch_bounds__(512) void cov_kernel(const float* __restrict__ x,
                                                  float* __restrict__ sums,
                                                  float* __restrict__ S,
                                                  int nchunks) {
    __shared__ float tile[16 * STRW];   // 16 p-pairs = 32 p rows, 34.8 KB
    const int t = threadIdx.x;
    const int lane = t & 31;
    const int wave = t >> 5;            // 0..15
    const int g = wave >> 2;            // group 0..3
    const int mtile = wave & 3;         // row-tile of S within group
    const int ln = lane & 15;
    const int k0 = (lane >> 4) << 1;    // 0 for lanes 0-15, 2 for lanes 16-31
    const int rowoff = (lane >= 16) ? 8 : 0;
    const int c4 = t & 63;              // fixed channel-quad this thread loads
    const int pp0 = t >> 6;             // p-pair 0..7 (also handles pp0+8)

    v8f acc0 = {}, acc1 = {}, acc2 = {}, acc3 = {};
    float4 csum = make_float4(0.f, 0.f, 0.f, 0.f);

    const int cA = (g * 64 + mtile * 16 + ln) * 2;   // word offset of A column
    const int cB = (g * 64 + ln) * 2;                // word offset of B col base
    const int pairoff = k0 >> 1;                     // 0 or 1

    int chunk = blockIdx.x;
    float4 va0, vb0, va1, vb1;
    if (chunk < nchunks) {
        const float* src = x + chunk * (TPB * 256) + c4 * 4;
        va0 = *(const float4*)(src + (pp0 * 2     ) * 256);
        vb0 = *(const float4*)(src + (pp0 * 2 + 1 ) * 256);
        va1 = *(const float4*)(src + (pp0 * 2 + 16) * 256);
        vb1 = *(const float4*)(src + (pp0 * 2 + 17) * 256);
    }
    while (chunk < nchunks) {
        // interleaved b128 stores (2 rows zipped along c)
        float* d0 = &tile[(pp0    ) * STRW + c4 * 8];
        float* d1 = &tile[(pp0 + 8) * STRW + c4 * 8];
        *(float4*)(d0    ) = make_float4(va0.x, vb0.x, va0.y, vb0.y);
        *(float4*)(d0 + 4) = make_float4(va0.z, vb0.z, va0.w, vb0.w);
        *(float4*)(d1    ) = make_float4(va1.x, vb1.x, va1.y, vb1.y);
        *(float4*)(d1 + 4) = make_float4(va1.z, vb1.z, va1.w, vb1.w);
        csum.x += va0.x + vb0.x + va1.x + vb1.x;
        csum.y += va0.y + vb0.y + va1.y + vb1.y;
        csum.z += va0.z + vb0.z + va1.z + vb1.z;
        csum.w += va0.w + vb0.w + va1.w + vb1.w;
        __syncthreads();

        const int next = chunk + gridDim.x;
        if (next < nchunks) {           // preload next chunk: overlaps WMMA below
            const float* src = x + next * (TPB * 256) + c4 * 4;
            __builtin_prefetch(src + (size_t)gridDim.x * TPB * 256, 0, 1);
            va0 = *(const float4*)(src + (pp0 * 2     ) * 256);
            vb0 = *(const float4*)(src + (pp0 * 2 + 1 ) * 256);
            va1 = *(const float4*)(src + (pp0 * 2 + 16) * 256);
            vb1 = *(const float4*)(src + (pp0 * 2 + 17) * 256);
        }
#pragma unroll
        for (int ks = 0; ks < TPB / 4; ++ks) {
            const float* base = &tile[(ks * 2 + pairoff) * STRW];
            const v2f a  = *(const v2f*)(base + cA);
            const v2f b0 = *(const v2f*)(base + cB);
            const v2f b1 = *(const v2f*)(base + cB + 32);
            const v2f b2 = *(const v2f*)(base + cB + 64);
            const v2f b3 = *(const v2f*)(base + cB + 96);
            acc0 = wmma_f32(a, b0, acc0);
            acc1 = wmma_f32(a, b1, acc1);
            acc2 = wmma_f32(a, b2, acc2);
            acc3 = wmma_f32(a, b3, acc3);
        }
        __syncthreads();
        chunk = next;
    }
    // Flush per-channel partial sums (thread owns channels c4*4 .. c4*4+3).
    atomicAdd(&sums[c4 * 4 + 0], csum.x);
    atomicAdd(&sums[c4 * 4 + 1], csum.y);
    atomicAdd(&sums[c4 * 4 + 2], csum.z);
    atomicAdd(&sums[c4 * 4 + 3], csum.w);
    // Flush Gram accumulators: VGPR r of lane l is element (M = r + rowoff, N = ln).
    const int rowb = g * 64 + mtile * 16 + rowoff;
#pragma unroll
    for (int r = 0; r < 8; ++r) {
        const int row = rowb + r;
        atomicAdd(&S[row * 64 +  0 + ln], acc0[r]);
        atomicAdd(&S[row * 64 + 16 + ln], acc1[r]);
        atomicAdd(&S[row * 64 + 32 + ln], acc2[r]);
        atomicAdd(&S[row * 64 + 48 + ln], acc3[r]);
    }
}

// ---------------------------------------------------------------------------
// Kernel 2: finalize cov, Newton-Schulz iterations, emit wm and off = wm*mu
// One block per group, 16 waves; every 64x64 matmul is a 4x4 grid of WMMA tiles.
// ---------------------------------------------------------------------------
__device__ __forceinline__ v8f mm_tile64(const float* A, const float* B,
                                         int mt, int nt, int ln, int k0) {
    v8f acc = {};
#pragma unroll
    for (int ks = 0; ks < 16; ++ks) {
        const int kk = ks * 4 + k0;
        v2f a, b;
        a.x = A[(mt * 16 + ln) * LDSC + kk];
        a.y = A[(mt * 16 + ln) * LDSC + kk + 1];
        b.x = B[kk * LDSC + nt * 16 + ln];
        b.y = B[(kk + 1) * LDSC + nt * 16 + ln];
        acc = wmma_f32(a, b, acc);
    }
    return acc;
}

__global__ __launch_bounds__(512) void ns_kernel(const float* __restrict__ S,
                                                 const float* __restrict__ sums,
                                                 float* __restrict__ wm_out,
                                                 float* __restrict__ mu_out,
                                                 float* __restrict__ off_out,
                                                 float Nf) {
    __shared__ float sigma[64 * LDSC];
    __shared__ float P[64 * LDSC];
    __shared__ float T[64 * LDSC];
    __shared__ float mu[64];
    __shared__ float trace_s;
    const int g = blockIdx.x;
    const int t = threadIdx.x;
    if (t < 64) mu[t] = sums[g * 64 + t] / Nf;
    if (t == 0) trace_s = 0.f;
    __syncthreads();
    // cov = (S - N*mu*mu^T)/(N-1), with shrinkage; P = I
    for (int i = t; i < 4096; i += 512) {
        const int m = i >> 6, k = i & 63;
        float c = (S[(g * 64 + m) * 64 + k] - Nf * mu[m] * mu[k]) / (Nf - 1.0f);
        c = (1.0f - EPSV) * c + ((m == k) ? EPSV : 0.0f);
        sigma[m * LDSC + k] = c;
        P[m * LDSC + k] = (m == k) ? 1.0f : 0.0f;
    }
    __syncthreads();
    if (t < 64) atomicAdd(&trace_s, sigma[t * LDSC + t]);   // ds_add_f32
    __syncthreads();
    const float trace = trace_s;
    for (int i = t; i < 4096; i += 512) {
        const int m = i >> 6, k = i & 63;
        sigma[m * LDSC + k] /= trace;
    }
    __syncthreads();

    const int lane = t & 31, wave = t >> 5;
    const int mt = wave >> 2, nt = wave & 3;
    const int ln = lane & 15, k0 = (lane >> 4) << 1;
    const int rowoff = (lane >= 16) ? 8 : 0;

    // P = (3P - P*P*P*sigma)/2, right-associated: T1=P@sigma; T2=P@T1; T3=P@T2
    for (int it = 0; it < 5; ++it) {
        v8f t1 = mm_tile64(P, sigma, mt, nt, ln, k0);
#pragma unroll
        for (int r = 0; r < 8; ++r)
            T[(mt * 16 + r + rowoff) * LDSC + nt * 16 + ln] = t1[r];
        __syncthreads();
        v8f t2 = mm_tile64(P, T, mt, nt, ln, k0);
        __syncthreads();                       // all reads of T done before overwrite
#pragma unroll
        for (int r = 0; r < 8; ++r)
            T[(mt * 16 + r + rowoff) * LDSC + nt * 16 + ln] = t2[r];
        __syncthreads();
        v8f t3 = mm_tile64(P, T, mt, nt, ln, k0);
        __syncthreads();                       // all reads of P done before update
#pragma unroll
        for (int r = 0; r < 8; ++r) {
            const int idx = (mt * 16 + r + rowoff) * LDSC + nt * 16 + ln;
            P[idx] = 1.5f * P[idx] - 0.5f * t3[r];
        }
        __syncthreads();
    }
    const float rs = rsqrtf(trace);
    for (int i = t; i < 4096; i += 512) {
        const int m = i >> 6, k = i & 63;
        wm_out[(g * 64 + m) * 64 + k] = P[m * LDSC + k] * rs;
    }
    if (t < 64) {
        mu_out[g * 64 + t] = mu[t];
        float o = 0.f;
#pragma unroll 8
        for (int k = 0; k < 64; ++k) o += P[t * LDSC + k] * rs * mu[k];
        off_out[g * 64 + t] = o;               // off = wm @ mu (absorbs centering)
    }
}

// ---------------------------------------------------------------------------
// Kernel 3: out = gamma * (wm @ x - off) + beta, per (p-chunk, group) block
// Row-major [p][c] tile: WMMA operand pairs (k,k+1) are contiguous b64 loads.
// ---------------------------------------------------------------------------
__global__ __launch_bounds__(512) void whiten_kernel(const float* __restrict__ x,
                                                     const float* __restrict__ wm,
                                                     const float* __restrict__ off,
                                                     const float* __restrict__ gamma,
                                                     const float* __restrict__ beta,
                                                     float* __restrict__ out,
                                                     int nchunks) {
    __shared__ float wmg[64 * LDSC];
    __shared__ float tile[64 * LDSC];
    __shared__ float og[64], gg[64], bb[64];
    const int g = blockIdx.y;
    const int t = threadIdx.x;
    for (int i = t; i < 4096; i += 512) {
        const int m = i >> 6, k = i & 63;
        wmg[m * LDSC + k] = wm[(g * 64 + m) * 64 + k];
    }
    if (t < 64) {
        og[t] = off[g * 64 + t];
        gg[t] = gamma[g * 64 + t];
        bb[t] = beta[g * 64 + t];
    }

    const int lane = t & 31, wave = t >> 5;
    const int mt = wave >> 2, pt = wave & 3;
    const int ln = lane & 15, k0 = (lane >> 4) << 1;
    const int rowoff = (lane >= 16) ? 8 : 0;
    const int lrow = t >> 4;            // 0..31 (also lrow+32)
    const int lc4 = (t & 15) * 4;       // channel quad within group

    int chunk = blockIdx.x;
    float4 r0, r1;
    if (chunk < nchunks) {
        const float* src = x + chunk * (64 * 256) + g * 64 + lc4;
        r0 = *(const float4*)(src + (lrow     ) * 256);
        r1 = *(const float4*)(src + (lrow + 32) * 256);
    }
    __syncthreads();                    // wmg/og/gg/bb ready

    while (chunk < nchunks) {
        *(float4*)&tile[(lrow     ) * LDSC + lc4] = r0;
        *(float4*)&tile[(lrow + 32) * LDSC + lc4] = r1;
        __syncthreads();

        const int next = chunk + gridDim.x;
        if (next < nchunks) {           // preload: overlaps WMMA below
            const float* src = x + next * (64 * 256) + g * 64 + lc4;
            __builtin_prefetch(src + (size_t)gridDim.x * 64 * 256, 0, 1);
            r0 = *(const float4*)(src + (lrow     ) * 256);
            r1 = *(const float4*)(src + (lrow + 32) * 256);
        }

        v8f acc = {};
#pragma unroll
        for (int ks = 0; ks < 16; ++ks) {
            const int kk = ks * 4 + k0;
            const v2f a = *(const v2f*)&wmg[(mt * 16 + ln) * LDSC + kk];   // A[m][k]
            const v2f b = *(const v2f*)&tile[(pt * 16 + ln) * LDSC + kk];  // B[k][p]=x[p][k]
            acc = wmma_f32(a, b, acc);
        }
        // Store: lane=spatial p, VGPR r = channel; 8 contiguous channels per lane.
        const int cl = mt * 16 + rowoff;
        const int p = chunk * 64 + pt * 16 + ln;
        float4 o0, o1;
        o0.x = gg[cl + 0] * (acc[0] - og[cl + 0]) + bb[cl + 0];
        o0.y = gg[cl + 1] * (acc[1] - og[cl + 1]) + bb[cl + 1];
        o0.z = gg[cl + 2] * (acc[2] - og[cl + 2]) + bb[cl + 2];
        o0.w = gg[cl + 3] * (acc[3] - og[cl + 3]) + bb[cl + 3];
        o1.x = gg[cl + 4] * (acc[4] - og[cl + 4]) + bb[cl + 4];
        o1.y = gg[cl + 5] * (acc[5] - og[cl + 5]) + bb[cl + 5];
        o1.z = gg[cl + 6] * (acc[6] - og[cl + 6]) + bb[cl + 6];
        o1.w = gg[cl + 7] * (acc[7] - og[cl + 7]) + bb[cl + 7];
        float* dst = out + p * 256 + g * 64 + cl;
        *(float4*)dst = o0;
        *(float4*)(dst + 4) = o1;
        __syncthreads();                // tile reads done before next overwrite
        chunk = next;
    }
}

// ---------------------------------------------------------------------------
extern "C" void kernel_launch(void* const* d_in, const int* in_sizes, int n_in,
                              void* d_out, int out_size, void* d_ws, size_t ws_size,
                              hipStream_t stream) {
    const float* x     = (const float*)d_in[0];
    const float* gamma = (const float*)d_in[1];
    const float* beta  = (const float*)d_in[2];
    float* out = (float*)d_out;

    float* ws   = (float*)d_ws;
    float* sums = ws;                 // 256
    float* S    = ws + 256;           // 4*64*64 = 16384
    float* wm   = ws + 256 + 16384;   // 16384
    float* offv = wm + 16384;         // 256
    float* muv  = offv + 256;         // 256

    const int N = in_sizes[0] / 256;  // 200704 spatial positions

    hipMemsetAsync((void*)ws, 0, (256 + 16384) * sizeof(float), stream);
    cov_kernel<<<512, 512, 0, stream>>>(x, sums, S, N / TPB);
    ns_kernel<<<4, 512, 0, stream>>>(S, sums, wm, muv, offv, (float)N);
    whiten_kernel<<<dim3(256, 4), 512, 0, stream>>>(x, wm, offv, gamma, beta, out, N / 64);
}